// OpticalOLSAdapter_20263655703345
// MI455X (gfx1250) — compile-verified
//
#include <hip/hip_runtime.h>
#include <math.h>

typedef float v2f __attribute__((ext_vector_type(2)));
typedef float v8f __attribute__((ext_vector_type(8)));
typedef unsigned int u32x4 __attribute__((ext_vector_type(4)));
typedef int i32x8 __attribute__((ext_vector_type(8)));
typedef int i32x4 __attribute__((ext_vector_type(4)));

#define TWO_PI_F 6.2831853071795864769f

// geometry
#define NP    512          // padded FFT size
#define IMGN  256          // image H=W
#define PADO  128          // (512-256)/2
#define NIMG  128          // B*C = 32*4
#define DF_F  244.140625f  // 1/(512 * 8e-6)

#if defined(__has_builtin)
#if __has_builtin(__builtin_amdgcn_tensor_load_to_lds) && \
    __has_builtin(__builtin_amdgcn_s_wait_tensorcnt)
#define USE_TDM 1
#endif
#endif

__device__ __forceinline__ float2 cmulf(float2 a, float cr, float ci) {
  return make_float2(a.x * cr - a.y * ci, a.x * ci + a.y * cr);
}

// ---- WMMA f32 16x16x4: D = A * B + C (all modifiers must be 0 for f32) ----
static __device__ __forceinline__ v8f wmma_f32(v2f a, v2f b, v8f c) {
  return __builtin_amdgcn_wmma_f32_16x16x4_f32(false, a, false, b, (short)0, c,
                                               false, false);
}

// Forward W16 DFT matrix as A-fragments (16x4 chunks, K = 4c..4c+3), plus the
// negated imaginary part. Inverse DFT = conj -> swap (ai, ain) at call sites.
__device__ __forceinline__ void dft16_frags(int lane, v2f* ar, v2f* ai, v2f* ain) {
  int m = lane & 15;
  int kof = (lane < 16) ? 0 : 2;
#pragma unroll
  for (int c = 0; c < 4; ++c) {
    float k0 = (float)(4 * c + kof);
    float a0 = -(TWO_PI_F / 16.0f) * (float)m * k0;
    float a1 = -(TWO_PI_F / 16.0f) * (float)m * (k0 + 1.0f);
    float s0, c0, s1, c1;
    __sincosf(a0, &s0, &c0);
    __sincosf(a1, &s1, &c1);
    v2f r, i, in;
    r.x = c0;  r.y = c1;
    i.x = s0;  i.y = s1;
    in.x = -s0; in.y = -s1;
    ar[c] = r; ai[c] = i; ain[c] = in;
  }
}

// B fragment (4x16) for chunk: element (K,N) = src[K*32 + ncol0 + N]
__device__ __forceinline__ void load_bfrag(const float2* src, int chunk, int ncol0,
                                           int lane, v2f& br, v2f& bi) {
  int nn = lane & 15;
  int kof = (lane < 16) ? 0 : 2;
  int K = 4 * chunk + kof;
  float2 e0 = src[K * 32 + ncol0 + nn];
  float2 e1 = src[(K + 1) * 32 + ncol0 + nn];
  br.x = e0.x; br.y = e1.x;
  bi.x = e0.y; bi.y = e1.y;
}

// complex 16x16 <- W16(16x16) * X(16x16 tile at ncol0), accumulate.
__device__ __forceinline__ void cwmma(const v2f* ar, const v2f* ai, const v2f* ain,
                                      const float2* src, int ncol0, int lane,
                                      v8f& yr, v8f& yi) {
#pragma unroll
  for (int c = 0; c < 4; ++c) {
    v2f br, bi;
    load_bfrag(src, c, ncol0, lane, br, bi);
    yr = wmma_f32(ar[c], br, yr);
    yr = wmma_f32(ain[c], bi, yr);
    yi = wmma_f32(ar[c], bi, yi);
    yi = wmma_f32(ai[c], br, yi);
  }
}

// 512-point complex FFT per wave in LDS (buf in/out, tmp scratch).
// N = 16*32 Cooley-Tukey, 32 = 2*16; DFT-16s via WMMA.
// Inverse FFT: sgn=+1 and pass conjugated fragments (ar, ain, ai).
// Tiles are processed sequentially: tile t's writeback (cols 16t..16t+15 in
// stage 1; scattered rows in stage 3) never aliases the other tile's reads,
// halving live accumulator registers.
__device__ void fft512(float2* buf, float2* tmp, int lane, float sgn, float scale,
                       const v2f* ar, const v2f* ai, const v2f* ain) {
  const int n = lane & 15;
  const int mof = (lane < 16) ? 0 : 8;

  // -- Stage 1: Y(16x32) = W16 * X(16x32); twiddle W512^(m*n2); in place --
#pragma unroll
  for (int tt = 0; tt < 2; ++tt) {
    v8f yr = {0.f,0.f,0.f,0.f,0.f,0.f,0.f,0.f};
    v8f yi = yr;
    cwmma(ar, ai, ain, buf, 16 * tt, lane, yr, yi);
    int n2 = 16 * tt + n;
    // twiddle recurrence along m: w(m) = exp(i*th*m), m = mof..mof+7
    float th = sgn * (TWO_PI_F / 512.0f) * (float)n2;
    float sst, cst; __sincosf(th, &sst, &cst);                 // step
    float s0, c0;  __sincosf(th * (float)mof, &s0, &c0);       // start
#pragma unroll
    for (int r = 0; r < 8; ++r) {
      int m = r + mof;
      buf[m * 32 + n2] = make_float2(yr[r] * c0 - yi[r] * s0,
                                     yr[r] * s0 + yi[r] * c0);
      float cn = c0 * cst - s0 * sst;
      s0 = c0 * sst + s0 * cst;
      c0 = cn;
    }
  }

  // -- Stage 2: radix-2 over the two 16-halves of n2; twiddle W32^t; buf->tmp
  //    t = lane&15 is loop-invariant -> single sincos.
  {
    int tq = lane & 15;
    int kb = lane >> 4;                  // 0 or 1
    float ang = sgn * (TWO_PI_F / 32.0f) * (float)tq;
    float s, c; __sincosf(ang, &s, &c);
#pragma unroll
    for (int j = 0; j < 8; ++j) {
      int k1 = kb + 2 * j;
      float2 u = buf[k1 * 32 + tq];
      float2 v = buf[k1 * 32 + 16 + tq];
      float2 h0 = make_float2(u.x + v.x, u.y + v.y);
      float2 d  = make_float2(u.x - v.x, u.y - v.y);
      tmp[tq * 32 + 2 * k1 + 0] = h0;
      tmp[tq * 32 + 2 * k1 + 1] = cmulf(d, c, s);
    }
  }

  // -- Stage 3: R(16x32) = W16 * Z(16x32); scatter X[k1 + 16*k2a + 32*k2b] --
#pragma unroll
  for (int tt = 0; tt < 2; ++tt) {
    v8f zr = {0.f,0.f,0.f,0.f,0.f,0.f,0.f,0.f};
    v8f zi = zr;
    cwmma(ar, ai, ain, tmp, 16 * tt, lane, zr, zi);
    int col = 16 * tt + n;
    int obase = (col >> 1) + 16 * (col & 1);
#pragma unroll
    for (int r = 0; r < 8; ++r) {
      int m = r + mof;                   // k2b
      buf[obase + 32 * m] = make_float2(scale * zr[r], scale * zi[r]);
    }
  }
}

// ---------------------------------------------------------------------------
// TDM: 1-D tile (512 x 8B) global -> LDS, descriptor per ISA chapter 8.
// ldsByteOff must be the byte offset of the destination inside the block's
// LDS allocation (single __shared__ array => base offset 0).
__device__ __forceinline__ void tdm_load_row(const float2* gsrc, unsigned ldsByteOff) {
#ifdef USE_TDM
  unsigned long long ga = (unsigned long long)(uintptr_t)gsrc;
  unsigned glo = (unsigned)__builtin_amdgcn_readfirstlane((int)(unsigned)(ga & 0xffffffffull));
  unsigned ghi = (unsigned)__builtin_amdgcn_readfirstlane((int)(unsigned)(ga >> 32));
  unsigned lof = (unsigned)__builtin_amdgcn_readfirstlane((int)ldsByteOff);
  u32x4 g0;
  g0.x = 1u;                                   // count=1, user descriptor
  g0.y = lof;                                  // lds_addr
  g0.z = glo;                                  // global_addr[31:0]
  g0.w = (ghi & 0x01ffffffu) | (2u << 30);     // global_addr[56:32] | type=2
  i32x8 g1;
  g1[0] = 0x00030000;                          // data_size=3 (8B), mask=0
  g1[1] = (int)(512u << 16);                   // tensor_dim0 = 512 (lo16)
  g1[2] = (int)(1u << 16);                     // tensor_dim1 = 1
  g1[3] = (int)(512u << 16);                   // tile_dim0 = 512
  g1[4] = 0;                                   // tile_dim1/2 unused
  g1[5] = 512;                                 // tensor_dim0_stride = 512
  g1[6] = 0;
  g1[7] = 0;
  i32x4 gz4 = {0, 0, 0, 0};
  i32x8 gz8 = {0, 0, 0, 0, 0, 0, 0, 0};
  __builtin_amdgcn_tensor_load_to_lds(g0, g1, gz4, gz4, gz8, 0);
#else
  (void)gsrc; (void)ldsByteOff;
#endif
}

__device__ __forceinline__ void tdm_wait() {
#ifdef USE_TDM
  __builtin_amdgcn_s_wait_tensorcnt(0);
#endif
}

// ---------------------------------------------------------------------------
// Kernel 1: clamp(z_map,0) -> pad -> forward row FFT. 4 rows per block.
__global__ void k_row_first(const float* __restrict__ zmap, float2* __restrict__ field) {
  __shared__ float2 bufA[4][512];
  __shared__ float2 bufB[4][512];
  int wave = threadIdx.x >> 5, lane = threadIdx.x & 31;
  int grow = blockIdx.x * 4 + wave;       // [0, 128*512)
  int img = grow >> 9;
  int row = grow & 511;
  float2* out = field + ((long)img << 18) + ((long)row << 9);
  if (row < PADO || row >= PADO + IMGN) {
    for (int j = lane; j < 512; j += 32) out[j] = make_float2(0.f, 0.f);
    return;
  }
  const float* zr = zmap + (long)img * IMGN * IMGN + (long)(row - PADO) * IMGN;
  __builtin_prefetch(zr, 0, 1);
  float2* A = bufA[wave];
  float2* T = bufB[wave];
  for (int j = lane; j < 512; j += 32) {
    float re = 0.f;
    if (j >= PADO && j < PADO + IMGN) re = fmaxf(zr[j - PADO], 0.f);
    A[j] = make_float2(re, 0.f);
  }
  v2f ar[4], ai[4], ain[4];
  dft16_frags(lane, ar, ai, ain);
  fft512(A, T, lane, -1.f, 1.f, ar, ai, ain);
  for (int j = lane; j < 512; j += 32) out[j] = A[j];
}

// ---------------------------------------------------------------------------
// Kernel 2: column pass. fwd col FFT -> *Hf(z) (band-limited ASM transfer) ->
// inverse col FFT. 8-column tiles, 2 columns per wave.
__device__ __forceinline__ float fidx(int i) {
  return (float)((i < 256) ? i : i - 512);
}

__global__ void k_col(float2* __restrict__ field, float kzScale, float fLim) {
  __shared__ float2 tile[8][514];   // padded stride: banks spread
  __shared__ float2 tmp[4][512];
  int t = threadIdx.x;
  int wave = t >> 5, lane = t & 31;
  int img = blockIdx.x >> 6;        // 64 tiles / image
  int cb = (blockIdx.x & 63) << 3;  // column base
  float2* base = field + ((long)img << 18);

  int c = t & 7, r0 = t >> 3;       // 16 row-groups
  for (int r = r0; r < 512; r += 16)
    tile[c][r] = base[(long)r * 512 + cb + c];
  __syncthreads();

  v2f ar[4], ai[4], ain[4];
  dft16_frags(lane, ar, ai, ain);

#pragma unroll
  for (int cc = 0; cc < 2; ++cc)
    fft512(&tile[wave * 2 + cc][0], tmp[wave], lane, -1.f, 1.f, ar, ai, ain);

  // transfer function Hf with Matsushima band-limit
#pragma unroll
  for (int cc = 0; cc < 2; ++cc) {
    int col = wave * 2 + cc;
    float fx = fidx(cb + col) * DF_F;
    float lfx = 532.0e-9f * fx;
    for (int r = lane; r < 512; r += 32) {
      float fy = fidx(r) * DF_F;
      float lfy = 532.0e-9f * fy;
      float arg = 1.f - lfx * lfx - lfy * lfy;
      float2 e = tile[col][r];
      if (arg > 0.f && fabsf(fx) <= fLim && fabsf(fy) <= fLim) {
        float ph = kzScale * sqrtf(arg);
        float s, co; __sincosf(ph, &s, &co);
        e = cmulf(e, co, s);
      } else {
        e = make_float2(0.f, 0.f);
      }
      tile[col][r] = e;
    }
  }

#pragma unroll
  for (int cc = 0; cc < 2; ++cc)  // inverse: conjugate frags (swap ai/ain)
    fft512(&tile[wave * 2 + cc][0], tmp[wave], lane, +1.f, 1.0f / 512.0f, ar, ain, ai);

  __syncthreads();
  for (int r = r0; r < 512; r += 16)
    base[(long)r * 512 + cb + c] = tile[c][r];
}

// ---------------------------------------------------------------------------
// Kernel 3: inverse row FFT -> crop/zero + *exp(i*phase) -> forward row FFT.
// Row loads via Tensor Data Mover (1 TDM op per wave).
__global__ void k_maskfft(float2* __restrict__ field, const float* __restrict__ mask) {
  __shared__ float2 lds[4096];      // [0..2047]: A rows, [2048..4095]: scratch
  int wave = threadIdx.x >> 5, lane = threadIdx.x & 31;
  int grow = blockIdx.x * 4 + wave;
  int img = grow >> 9;
  int row = grow & 511;
  float2* io = field + ((long)img << 18) + ((long)row << 9);
  if (row < PADO || row >= PADO + IMGN) {  // masked to zero anyway
    for (int j = lane; j < 512; j += 32) io[j] = make_float2(0.f, 0.f);
    return;
  }
  float2* A = &lds[wave * 512];
  float2* T = &lds[2048 + wave * 512];

#ifdef USE_TDM
  tdm_load_row(io, (unsigned)(wave * 512u * sizeof(float2)));
#endif

  v2f ar[4], ai[4], ain[4];
  dft16_frags(lane, ar, ai, ain);

#ifdef USE_TDM
  tdm_wait();
#else
  for (int j = lane; j < 512; j += 32) A[j] = io[j];
#endif

  fft512(A, T, lane, +1.f, 1.0f / 512.0f, ar, ain, ai);   // to spatial

  const float* mrow = mask + (long)(row - PADO) * IMGN;
  for (int j = lane; j < 512; j += 32) {
    float2 e = A[j];
    if (j >= PADO && j < PADO + IMGN) {
      float s, c; __sincosf(mrow[j - PADO], &s, &c);
      e = cmulf(e, c, s);
    } else {
      e = make_float2(0.f, 0.f);
    }
    A[j] = e;
  }

  fft512(A, T, lane, -1.f, 1.f, ar, ai, ain);             // back to freq (x)
  for (int j = lane; j < 512; j += 32) io[j] = A[j];
}

// ---------------------------------------------------------------------------
// Kernel 4: inverse row FFT of interior rows -> |E|^2 -> 2x2 avg pool -> out.
// 4 interior rows per block => 2 output rows. TDM row loads.
__global__ void k_final(const float2* __restrict__ field, float* __restrict__ out) {
  __shared__ float2 lds[4096];      // [0..2047]: A rows, [2048..4095]: scratch
  int wave = threadIdx.x >> 5, lane = threadIdx.x & 31;
  int gr = blockIdx.x * 4 + wave;     // interior-row id in [0, 128*256)
  int img = gr >> 8;
  int ri = gr & 255;
  int row = PADO + ri;
  const float2* in = field + ((long)img << 18) + ((long)row << 9);
  float2* A = &lds[wave * 512];
  float2* T = &lds[2048 + wave * 512];

#ifdef USE_TDM
  tdm_load_row(in, (unsigned)(wave * 512u * sizeof(float2)));
#endif

  v2f ar[4], ai[4], ain[4];
  dft16_frags(lane, ar, ai, ain);

#ifdef USE_TDM
  tdm_wait();
#else
  for (int j = lane; j < 512; j += 32) A[j] = in[j];
#endif

  fft512(A, T, lane, +1.f, 1.0f / 512.0f, ar, ain, ai);

  // intensity into this wave's scratch region (free after fft512)
  float* inten = (float*)T;
  for (int j = lane; j < 256; j += 32) {
    float2 e = A[PADO + j];
    inten[j] = e.x * e.x + e.y * e.y;
  }
  __syncthreads();

  int ribase = (blockIdx.x * 4) & 255;  // 4 | 256 -> block stays in one image
  int ox = threadIdx.x;                  // 0..127
  if (ox < 128) {
    const float* i0 = (const float*)&lds[2048];
#pragma unroll
    for (int half = 0; half < 2; ++half) {
      int oy = (ribase >> 1) + half;
      const float* ra = i0 + (2 * half) * 1024;      // wave stride = 512 f2 = 1024 f
      const float* rb = i0 + (2 * half + 1) * 1024;
      float v = 0.25f * (ra[2 * ox] + ra[2 * ox + 1] + rb[2 * ox] + rb[2 * ox + 1]);
      out[(long)img * (128 * 128) + (long)oy * 128 + ox] = fmaxf(v, 0.f);
    }
  }
}

// ---------------------------------------------------------------------------
extern "C" void kernel_launch(void* const* d_in, const int* in_sizes, int n_in,
                              void* d_out, int out_size, void* d_ws, size_t ws_size,
                              hipStream_t stream) {
  const float* zmap    = (const float*)d_in[0];
  const float* phase0  = (const float*)d_in[1];
  const float* phase1  = (const float*)d_in[2];
  const float* scatter = (const float*)d_in[3];
  float* outp = (float*)d_out;
  float2* field = (float2*)d_ws;     // 128 * 512*512 * 8B = 256 MB

  const double lam = 532.0e-9, pitch = 8.0e-6;
  const double df = 1.0 / (512.0 * pitch);
  const double z1 = 50.0e-3, z2 = 100.0e-3, zs = 1.0e-3;
  auto kz = [&](double z) { return (float)(2.0 * M_PI * z / lam); };
  auto fl = [&](double z) {
    double a = 2.0 * df * z;
    return (float)(1.0 / (lam * sqrt(a * a + 1.0)));
  };

  dim3 blk(128, 1, 1);
  k_row_first<<<16384, blk, 0, stream>>>(zmap, field);

  k_col<<<8192, blk, 0, stream>>>(field, kz(z1), fl(z1));
  k_maskfft<<<16384, blk, 0, stream>>>(field, phase0);

  k_col<<<8192, blk, 0, stream>>>(field, kz(z1), fl(z1));
  k_maskfft<<<16384, blk, 0, stream>>>(field, phase1);

  k_col<<<8192, blk, 0, stream>>>(field, kz(z2), fl(z2));
  k_maskfft<<<16384, blk, 0, stream>>>(field, scatter);

  k_col<<<8192, blk, 0, stream>>>(field, kz(zs), fl(zs));
  k_final<<<8192, blk, 0, stream>>>(field, outp);
}